// Net_65798898975397
// MI455X (gfx1250) — compile-verified
//
#include <hip/hip_runtime.h>

// Vanilla tanh-RNN scan, one wave32 per 16-batch tile, state kept in VGPRs.
// Recurrence matmul via V_WMMA_F32_16X16X4_F32, computing D' = W_hh x H^T so
// the A operand (W_hh) is loop-invariant and the state lives in the B layout.
//
// Zero-shuffle recurrence: in both the D layout and the B layout, the hidden
// index depends only on (register, lane-half) while the lane-within-half
// always carries the batch index. So the post-tanh D registers are fed
// DIRECTLY as the next step's B operand; the resulting fixed K-permutation
// pi is absorbed into the (loop-invariant) A operand by loading W_hh with
// pi-permuted columns once at setup. No cross-lane data movement in the loop
// except one half-swap (v_permlanex16_b32) for the output-head reduction.
//
// K=16 split into two independent 2-deep WMMA chains; tanh is the native
// CDNA5 v_tanh_f32 transcendental.
//
// Layout assumptions (CDNA5 ISA, 32-bit 16x16x4):
//   A (16x4):  lanes 0-15 / 16-31 hold M=0..15; VGPR j: K=j (lo half),
//              K=j+2 (hi half).
//   B (4x16):  symmetric: VGPR j holds K=j (lanes 0-15) / K=j+2 (lanes 16-31),
//              N = lane%16.
//   C/D:       VGPR r holds M-row r (lanes 0-15) / r+8 (lanes 16-31),
//              N = lane%16.  Here M = hidden index n, N = batch index m.
//   => pi(k) = 2*(k>>2) + (k&1) + 8*((k>>1)&1).

typedef __attribute__((ext_vector_type(2))) float v2f;
typedef __attribute__((ext_vector_type(8))) float v8f;

namespace {
constexpr int kB = 2048;
constexpr int kT = 1024;
constexpr int kH = 16;
// Hidden index held by hardware-K slot k when D registers are fed as B.
__device__ constexpr int kPi[16] = {0, 1, 8, 9, 2, 3, 10, 11,
                                    4, 5, 12, 13, 6, 7, 14, 15};
}

__device__ __forceinline__ float fast_tanh(float v) {
#if __has_builtin(__builtin_amdgcn_tanhf)
    return __builtin_amdgcn_tanhf(v);                 // v_tanh_f32 (TRANS32)
#else
    const float m = v * 2.8853900817779268f;          // 2/ln(2)
    const float e = __builtin_amdgcn_exp2f(m);        // v_exp_f32
    const float r = __builtin_amdgcn_rcpf(e + 1.0f);  // v_rcp_f32
    const float t = fmaf(-2.0f, r, 1.0f);             // 1 - 2/(e^{2v}+1)
    const float x2 = v * v;
    const float p  = v * fmaf(x2, -0.3333333433f, 1.0f);  // x - x^3/3
    return (__builtin_fabsf(v) < 0.04f) ? p : t;
#endif
}

// Swap data between the two 16-lane halves (shfl_xor 16) via
// v_permlanex16_b32 with identity nibble selectors. VALU-only.
__device__ __forceinline__ float halfswap(float v) {
#if __has_builtin(__builtin_amdgcn_permlanex16)
    int i = __builtin_bit_cast(int, v);
    i = __builtin_amdgcn_permlanex16(i, i, 0x76543210, 0xfedcba98, false, false);
    return __builtin_bit_cast(float, i);
#else
    return __shfl_xor(v, 16, 32);
#endif
}

__global__ __launch_bounds__(32) void rnn_scan_wmma(
    const float* __restrict__ x,      // [B, T] (I == 1)
    const float* __restrict__ h0,     // [1, B, H]
    const float* __restrict__ w_ih,   // [H, 1]
    const float* __restrict__ w_hh,   // [H, H]
    const float* __restrict__ b_ih,   // [H]
    const float* __restrict__ b_hh,   // [H]
    const float* __restrict__ w_lin,  // [1, H]
    const float* __restrict__ b_lin,  // [1]
    float* __restrict__ out)          // [B*T] ++ [B*H]
{
    const int lane = threadIdx.x & 31;
    const int m    = lane & 15;   // batch index within tile / A-row index
    const int hi   = lane >> 4;   // lane half
    const int row  = blockIdx.x * 16 + m;

    // A = W_hh with pi-permuted columns, in A-layout chunks (loop-invariant):
    // aw[2c+j] holds w_hh[n'][ pi(4c + j + 2*hi) ] with n' = lane%16.
    float aw[8];
#pragma unroll
    for (int c = 0; c < 4; ++c)
#pragma unroll
        for (int j = 0; j < 2; ++j)
            aw[2 * c + j] = w_hh[m * kH + kPi[4 * c + j + 2 * hi]];

    // Per-lane constants along the hidden axis: n = r + 8*hi (state/D layout).
    float wih_l[8], bias_l[8], wlin_l[8];
#pragma unroll
    for (int r = 0; r < 8; ++r) {
        const int n = r + 8 * hi;
        wih_l[r]  = w_ih[n];
        bias_l[r] = b_ih[n] + b_hh[n];
        wlin_l[r] = w_lin[n];
    }
    const float blin = b_lin[0];

    // Recurrent state: hs[r] holds h[m][n], n = r + 8*hi (uniform convention
    // for h0 load, WMMA C seed, tanh output, B-operand feed, h_T store).
    float hs[8];
#pragma unroll
    for (int r = 0; r < 8; ++r)
        hs[r] = h0[row * kH + r + 8 * hi];

    const float* xrow = x + (size_t)row * kT;
    float*       orow = out + (size_t)row * kT;
    const bool   lo   = (lane < 16);

    for (int t4 = 0; t4 < kT / 4; ++t4) {
        const float4 xq = *(const float4*)(xrow + 4 * t4);
        const float xs[4] = {xq.x, xq.y, xq.z, xq.w};
        float ov[4];
#pragma unroll
        for (int u = 0; u < 4; ++u) {
            // acc0 seeded with the input/bias term; acc1 seeded with zero.
            v8f acc0, acc1 = {};
#pragma unroll
            for (int r = 0; r < 8; ++r)
                acc0[r] = fmaf(xs[u], wih_l[r], bias_l[r]);

            // Two independent 2-deep WMMA chains over K=16; the B operand is
            // the previous step's D registers fed directly (pi absorbed in A).
            {
                v2f a0; a0[0] = aw[0]; a0[1] = aw[1];
                v2f b0; b0[0] = hs[0]; b0[1] = hs[1];
                v2f a1; a1[0] = aw[2]; a1[1] = aw[3];
                v2f b1; b1[0] = hs[2]; b1[1] = hs[3];
                acc0 = __builtin_amdgcn_wmma_f32_16x16x4_f32(
                    false, a0, false, b0, (short)0, acc0, false, false);
                acc1 = __builtin_amdgcn_wmma_f32_16x16x4_f32(
                    false, a1, false, b1, (short)0, acc1, false, false);
            }
            {
                v2f a2; a2[0] = aw[4]; a2[1] = aw[5];
                v2f b2; b2[0] = hs[4]; b2[1] = hs[5];
                v2f a3; a3[0] = aw[6]; a3[1] = aw[7];
                v2f b3; b3[0] = hs[6]; b3[1] = hs[7];
                acc0 = __builtin_amdgcn_wmma_f32_16x16x4_f32(
                    false, a2, false, b2, (short)0, acc0, false, false);
                acc1 = __builtin_amdgcn_wmma_f32_16x16x4_f32(
                    false, a3, false, b3, (short)0, acc1, false, false);
            }

            // New state (D layout == our state convention). No rearrange.
#pragma unroll
            for (int r = 0; r < 8; ++r)
                hs[r] = fast_tanh(acc0[r] + acc1[r]);

            // Output head: out[row*T + t] = sum_n h[m][n]*w_lin[n] + b_lin.
            float p = 0.0f;
#pragma unroll
            for (int r = 0; r < 8; ++r)
                p = fmaf(hs[r], wlin_l[r], p);
            p += halfswap(p);  // combine the two hidden halves
            ov[u] = p + blin;
        }
        if (lo) {
            float4 o4; o4.x = ov[0]; o4.y = ov[1]; o4.z = ov[2]; o4.w = ov[3];
            *(float4*)(orow + 4 * t4) = o4;
        }
    }

    // Final hidden state h_T: out[B*T + b*H + n], n = r + 8*hi.
#pragma unroll
    for (int r = 0; r < 8; ++r)
        out[(size_t)kB * kT + (size_t)row * kH + r + 8 * hi] = hs[r];
}

extern "C" void kernel_launch(void* const* d_in, const int* in_sizes, int n_in,
                              void* d_out, int out_size, void* d_ws, size_t ws_size,
                              hipStream_t stream) {
    (void)in_sizes; (void)n_in; (void)out_size; (void)d_ws; (void)ws_size;
    const float* x     = (const float*)d_in[0];
    const float* h0    = (const float*)d_in[1];
    const float* w_ih  = (const float*)d_in[2];
    const float* w_hh  = (const float*)d_in[3];
    const float* b_ih  = (const float*)d_in[4];
    const float* b_hh  = (const float*)d_in[5];
    const float* w_lin = (const float*)d_in[6];
    const float* b_lin = (const float*)d_in[7];
    float* out = (float*)d_out;

    rnn_scan_wmma<<<kB / 16, 32, 0, stream>>>(
        x, h0, w_ih, w_hh, b_ih, b_hh, w_lin, b_lin, out);
}